// OptimizedGraphAttentionNet_4423816314890
// MI455X (gfx1250) — compile-verified
//
#include <hip/hip_runtime.h>
#include <hip/hip_bf16.h>
#include <math.h>

#define N_NODES 50000
#define N_EDGES 800000
#define HEADS   4

typedef __attribute__((ext_vector_type(2))) float v2f;
typedef __attribute__((ext_vector_type(8))) float v8f;

// ---------- helpers ----------
__device__ __forceinline__ unsigned f2ord(float f) {
  unsigned u = __float_as_uint(f);
  return (u & 0x80000000u) ? ~u : (u | 0x80000000u);
}
__device__ __forceinline__ float ord2f(unsigned u) {
  u = (u & 0x80000000u) ? (u & 0x7fffffffu) : ~u;
  return __uint_as_float(u);
}
__device__ __forceinline__ void atomicAddF(float* p, float v) {
  __hip_atomic_fetch_add(p, v, __ATOMIC_RELAXED, __HIP_MEMORY_SCOPE_AGENT);
}

// ---------- fill ----------
__global__ void fill_u32(unsigned* __restrict__ p, unsigned v, long n) {
  long i = (long)blockIdx.x * blockDim.x + threadIdx.x;
  if (i < n) p[i] = v;
}

// ---------- WMMA fp32 GEMM: C = act(A[M,K] @ B[K,N] (+bias)) ----------
// block = 128 threads = 4 waves; each wave computes one 16x16 tile.
// grid.x = M/16, grid.y = ceil(N/64). M must be multiple of 16, K of 4, N of 16.
template <int ACT>
__global__ void wmma_gemm(const float* __restrict__ A, const float* __restrict__ B,
                          const float* __restrict__ bias, float* __restrict__ C,
                          int M, int N, int K) {
  const int wave  = threadIdx.x >> 5;
  const int lane  = threadIdx.x & 31;
  const int lmod  = lane & 15;
  const int lhalf = lane >> 4;
  const int row0  = blockIdx.x * 16;
  const int col0  = (blockIdx.y * 4 + wave) * 16;
  if (col0 >= N) return;
  const int arow = row0 + lmod;
  const int bcol = col0 + lmod;
  v8f acc = {};
  for (int k = 0; k < K; k += 4) {
    const int ka = k + lhalf * 2;
    // A fragment: lane<16 -> (M=lmod, K=ka..ka+1), lane>=16 -> K=ka+2.. handled by lhalf
    const float2 av = *(const float2*)(A + (size_t)arow * K + ka);
    v2f a; a.x = av.x; a.y = av.y;
    v2f b;
    b.x = B[(size_t)ka * N + bcol];
    b.y = B[(size_t)(ka + 1) * N + bcol];
    acc = __builtin_amdgcn_wmma_f32_16x16x4_f32(false, a, false, b,
                                                (short)0, acc, false, false);
  }
  const float bv = bias ? bias[bcol] : 0.0f;
#pragma unroll
  for (int r = 0; r < 8; ++r) {
    float v = acc[r] + bv;
    if (ACT == 1) v = fmaxf(v, 0.0f);
    const int row = row0 + lhalf * 8 + r;
    C[(size_t)row * N + bcol] = v;
  }
}

// ---------- per-node attention scalars: a[n,h] = sum_c h[n,h,c]*att[h,c] ----------
__global__ void node_att(const float* __restrict__ h, const float* __restrict__ att_src,
                         const float* __restrict__ att_dst, float* __restrict__ asrc,
                         float* __restrict__ adst, int Nn, int H, int C) {
  int i = blockIdx.x * blockDim.x + threadIdx.x;
  if (i >= Nn * H) return;
  int n = i / H, hh = i - n * H;
  const float* hp = h + (size_t)n * H * C + (size_t)hh * C;
  const float* ws = att_src + hh * C;
  const float* wd = att_dst + hh * C;
  float s = 0.f, d = 0.f;
  for (int c = 0; c < C; ++c) { float v = hp[c]; s += v * ws[c]; d += v * wd[c]; }
  asrc[i] = s; adst[i] = d;
}

__device__ __forceinline__ float leaky(float x) { return x > 0.f ? x : 0.2f * x; }

// ---------- edge pass A: segment max of leaky-relu logits via ordered-uint atomicMax ----------
__global__ void edge_max(const int* __restrict__ src, const int* __restrict__ dst,
                         const float* __restrict__ asrc, const float* __restrict__ adst,
                         unsigned* __restrict__ mmax, int E, int H) {
  int i = blockIdx.x * blockDim.x + threadIdx.x;
  if (i >= E * H) return;
  int e = i / H, hh = i - e * H;
  int s = src[e], d = dst[e];
  float lg = leaky(asrc[s * H + hh] + adst[d * H + hh]);
  atomicMax(&mmax[d * H + hh], f2ord(lg));
}

// ---------- edge pass B: ev = exp(logit - max); segment sum ----------
__global__ void edge_expsum(const int* __restrict__ src, const int* __restrict__ dst,
                            const float* __restrict__ asrc, const float* __restrict__ adst,
                            const unsigned* __restrict__ mmax, float* __restrict__ ev,
                            float* __restrict__ ssum, int E, int H) {
  int i = blockIdx.x * blockDim.x + threadIdx.x;
  if (i >= E * H) return;
  int e = i / H, hh = i - e * H;
  int s = src[e], d = dst[e];
  float m = ord2f(mmax[d * H + hh]);
  if (!isfinite(m)) m = 0.f;
  float lg = leaky(asrc[s * H + hh] + adst[d * H + hh]);
  float x = expf(lg - m);
  ev[i] = x;
  atomicAddF(&ssum[d * H + hh], x);
}

// ---------- edge pass C (layer 1): one edge per 256-thread block, 4 heads x 64 ch ----------
__global__ void edge_aggr1(const int* __restrict__ src, const int* __restrict__ dst,
                           const float* __restrict__ h1, const float* __restrict__ ev,
                           const float* __restrict__ ssum, float* __restrict__ out) {
  int e = blockIdx.x;
  int t = threadIdx.x;          // 0..255 -> (head = t>>6, channel = t&63)
  int hh = t >> 6;
  int s = src[e], d = dst[e];
  float alpha = ev[(size_t)e * HEADS + hh] / (ssum[d * HEADS + hh] + 1e-16f);
  float v = alpha * h1[(size_t)s * 256 + t];
  atomicAddF(&out[(size_t)d * 256 + t], v);
}

// ---------- edge pass C (layer 2): 1 head, 32 channels ----------
__global__ void edge_aggr2(const int* __restrict__ src, const int* __restrict__ dst,
                           const float* __restrict__ h2, const float* __restrict__ ev,
                           const float* __restrict__ ssum, float* __restrict__ out, int E) {
  int idx = blockIdx.x * blockDim.x + threadIdx.x;
  if (idx >= E * 32) return;
  int e = idx >> 5, c = idx & 31;
  int s = src[e], d = dst[e];
  float alpha = ev[e] / (ssum[d] + 1e-16f);
  atomicAddF(&out[(size_t)d * 32 + c], alpha * h2[(size_t)s * 32 + c]);
}

// ---------- GraphNorm: streaming per-channel sum & sumsq ----------
#define NORM_ROWS 128
__global__ void norm_reduce(const float* __restrict__ h, const float* __restrict__ bias,
                            float* __restrict__ sum, float* __restrict__ sumsq, int Nn) {
  int c = threadIdx.x;                 // 0..255, one channel per thread (coalesced rows)
  int row0 = blockIdx.x * NORM_ROWS;
  float s = 0.f, q = 0.f;
  for (int r = 0; r < NORM_ROWS; ++r) {
    int row = row0 + r;
    if (row < Nn) { float v = h[(size_t)row * 256 + c] + bias[c]; s += v; q += v * v; }
  }
  atomicAddF(&sum[c], s);
  atomicAddF(&sumsq[c], q);
}

__global__ void norm_params(const float* __restrict__ sum, const float* __restrict__ sumsq,
                            const float* __restrict__ gn_ms, const float* __restrict__ gn_w,
                            float* __restrict__ msmu, float* __restrict__ scale, int Nn) {
  int c = threadIdx.x;
  float inv_n = 1.0f / (float)Nn;
  float mu = sum[c] * inv_n;
  float mm = mu * gn_ms[c];
  // var = E[(h - mm)^2] = E[h^2] - 2*mm*E[h] + mm^2
  float var = sumsq[c] * inv_n - 2.f * mm * mu + mm * mm;
  msmu[c]  = mm;
  scale[c] = gn_w[c] * rsqrtf(var + 1e-5f);
}

__global__ void norm_apply_elu(const float* __restrict__ h, const float* __restrict__ g1b,
                               const float* __restrict__ msmu, const float* __restrict__ scale,
                               const float* __restrict__ gnb, float* __restrict__ out, long total) {
  long i = (long)blockIdx.x * blockDim.x + threadIdx.x;
  if (i >= total) return;
  int c = (int)(i & 255);
  float v = (h[i] + g1b[c] - msmu[c]) * scale[c] + gnb[c];
  out[i] = v > 0.f ? v : expm1f(v);
}

// ---------- log_softmax over 32 classes: one wave32 per node ----------
__global__ void log_softmax32(const float* __restrict__ h, const float* __restrict__ bias,
                              float* __restrict__ out, int Nn) {
  int wave = threadIdx.x >> 5;
  int lane = threadIdx.x & 31;
  int row = blockIdx.x * (blockDim.x >> 5) + wave;
  if (row >= Nn) return;
  float v = h[(size_t)row * 32 + lane] + bias[lane];
  float m = v;
  for (int o = 16; o > 0; o >>= 1) m = fmaxf(m, __shfl_xor(m, o, 32));
  float e = expf(v - m);
  float s = e;
  for (int o = 16; o > 0; o >>= 1) s += __shfl_xor(s, o, 32);
  out[(size_t)row * 32 + lane] = (v - m) - logf(s);
}

// ---------- host launch ----------
extern "C" void kernel_launch(void* const* d_in, const int* in_sizes, int n_in,
                              void* d_out, int out_size, void* d_ws, size_t ws_size,
                              hipStream_t stream) {
  const float* x     = (const float*)d_in[0];
  const int*   ei    = (const int*)d_in[1];
  const int*   src   = ei;
  const int*   dstp  = ei + N_EDGES;
  const float* t1_w  = (const float*)d_in[2];
  const float* t1_b  = (const float*)d_in[3];
  const float* t2_w  = (const float*)d_in[4];
  const float* t2_b  = (const float*)d_in[5];
  const float* g1_w  = (const float*)d_in[6];
  const float* g1_as = (const float*)d_in[7];
  const float* g1_ad = (const float*)d_in[8];
  const float* g1_b  = (const float*)d_in[9];
  const float* gn_w  = (const float*)d_in[10];
  const float* gn_b  = (const float*)d_in[11];
  const float* gn_ms = (const float*)d_in[12];
  const float* g2_w  = (const float*)d_in[13];
  const float* g2_as = (const float*)d_in[14];
  const float* g2_ad = (const float*)d_in[15];
  const float* g2_b  = (const float*)d_in[16];
  float* out = (float*)d_out;

  // workspace layout
  char* w = (char*)d_ws;
  auto alloc = [&](size_t bytes) -> char* {
    char* p = w; w += (bytes + 255) & ~(size_t)255; return p;
  };
  float*    h128  = (float*)   alloc((size_t)N_NODES * 128 * 4);  // also reused as ev1
  float*    h64   = (float*)   alloc((size_t)N_NODES * 64 * 4);
  float*    h1    = (float*)   alloc((size_t)N_NODES * 256 * 4);  // also reused as h2in
  float*    a1s   = (float*)   alloc((size_t)N_NODES * HEADS * 4);
  float*    a1d   = (float*)   alloc((size_t)N_NODES * HEADS * 4);
  unsigned* mmax1 = (unsigned*)alloc((size_t)N_NODES * HEADS * 4);
  float*    ssum1 = (float*)   alloc((size_t)N_NODES * HEADS * 4);
  float*    gat1  = (float*)   alloc((size_t)N_NODES * 256 * 4);
  float*    csum  = (float*)   alloc(256 * 4);
  float*    csq   = (float*)   alloc(256 * 4);
  float*    msmu  = (float*)   alloc(256 * 4);
  float*    scale = (float*)   alloc(256 * 4);
  float*    h2    = (float*)   alloc((size_t)N_NODES * 32 * 4);
  float*    a2s   = (float*)   alloc((size_t)N_NODES * 4);
  float*    a2d   = (float*)   alloc((size_t)N_NODES * 4);
  unsigned* mmax2 = (unsigned*)alloc((size_t)N_NODES * 4);
  float*    ssum2 = (float*)   alloc((size_t)N_NODES * 4);
  float*    ev2   = (float*)   alloc((size_t)N_EDGES * 4);
  float*    gat2  = (float*)   alloc((size_t)N_NODES * 32 * 4);
  float*    ev1   = h128;   // GEMM1 output dead after GEMM2
  float*    h2in  = h1;     // h1 dead after edge_aggr1

  const unsigned ORD_NEG_INF = 0x007FFFFFu;   // f2ord(-inf)
  const int MT = N_NODES / 16;                // 3125, exact

  // 1) h128 = relu(x @ t1_w + t1_b)           [50000,256]x[256,128]
  wmma_gemm<1><<<dim3(MT, 2), 128, 0, stream>>>(x, t1_w, t1_b, h128, N_NODES, 128, 256);
  // 2) h64 = relu(h128 @ t2_w + t2_b)         [50000,128]x[128,64]
  wmma_gemm<1><<<dim3(MT, 1), 128, 0, stream>>>(h128, t2_w, t2_b, h64, N_NODES, 64, 128);
  // 3) h1 = h64 @ g1_w                        [50000,64]x[64,256]
  wmma_gemm<0><<<dim3(MT, 4), 128, 0, stream>>>(h64, g1_w, nullptr, h1, N_NODES, 256, 64);
  // 4) per-node attention scalars (4 heads)
  node_att<<<(N_NODES * HEADS + 255) / 256, 256, 0, stream>>>(h1, g1_as, g1_ad, a1s, a1d,
                                                              N_NODES, HEADS, 64);
  // 5) init segment buffers for layer 1
  fill_u32<<<(N_NODES * HEADS + 255) / 256, 256, 0, stream>>>(mmax1, ORD_NEG_INF, (long)N_NODES * HEADS);
  fill_u32<<<(N_NODES * HEADS + 255) / 256, 256, 0, stream>>>((unsigned*)ssum1, 0u, (long)N_NODES * HEADS);
  fill_u32<<<((long)N_NODES * 256 + 255) / 256, 256, 0, stream>>>((unsigned*)gat1, 0u, (long)N_NODES * 256);
  // 6) segment softmax + aggregate (layer 1)
  edge_max   <<<(N_EDGES * HEADS + 255) / 256, 256, 0, stream>>>(src, dstp, a1s, a1d, mmax1, N_EDGES, HEADS);
  edge_expsum<<<(N_EDGES * HEADS + 255) / 256, 256, 0, stream>>>(src, dstp, a1s, a1d, mmax1, ev1, ssum1,
                                                                 N_EDGES, HEADS);
  edge_aggr1 <<<N_EDGES, 256, 0, stream>>>(src, dstp, h1, ev1, ssum1, gat1);
  // 7) GraphNorm (+g1_b folded in) then ELU
  fill_u32<<<2, 256, 0, stream>>>((unsigned*)csum, 0u, 512);
  norm_reduce<<<(N_NODES + NORM_ROWS - 1) / NORM_ROWS, 256, 0, stream>>>(gat1, g1_b, csum, csq, N_NODES);
  norm_params<<<1, 256, 0, stream>>>(csum, csq, gn_ms, gn_w, msmu, scale, N_NODES);
  norm_apply_elu<<<((long)N_NODES * 256 + 255) / 256, 256, 0, stream>>>(gat1, g1_b, msmu, scale, gn_b,
                                                                        h2in, (long)N_NODES * 256);
  // 8) h2 = h2in @ g2_w                       [50000,256]x[256,32]
  wmma_gemm<0><<<dim3(MT, 1), 128, 0, stream>>>(h2in, g2_w, nullptr, h2, N_NODES, 32, 256);
  // 9) layer-2 attention (1 head, 32 ch)
  node_att<<<(N_NODES + 255) / 256, 256, 0, stream>>>(h2, g2_as, g2_ad, a2s, a2d, N_NODES, 1, 32);
  fill_u32<<<(N_NODES + 255) / 256, 256, 0, stream>>>(mmax2, ORD_NEG_INF, (long)N_NODES);
  fill_u32<<<(N_NODES + 255) / 256, 256, 0, stream>>>((unsigned*)ssum2, 0u, (long)N_NODES);
  fill_u32<<<((long)N_NODES * 32 + 255) / 256, 256, 0, stream>>>((unsigned*)gat2, 0u, (long)N_NODES * 32);
  edge_max   <<<(N_EDGES + 255) / 256, 256, 0, stream>>>(src, dstp, a2s, a2d, mmax2, N_EDGES, 1);
  edge_expsum<<<(N_EDGES + 255) / 256, 256, 0, stream>>>(src, dstp, a2s, a2d, mmax2, ev2, ssum2, N_EDGES, 1);
  edge_aggr2 <<<((long)N_EDGES * 32 + 255) / 256, 256, 0, stream>>>(src, dstp, h2, ev2, ssum2, gat2, N_EDGES);
  // 10) log_softmax (+g2_b) -> d_out, one wave per node
  log_softmax32<<<(N_NODES + 3) / 4, 128, 0, stream>>>(gat2, g2_b, out, N_NODES);
}